// CTCLayer_13383118095054
// MI455X (gfx1250) — compile-verified
//
#include <hip/hip_runtime.h>

// ---------------------------------------------------------------------------
// CTC batch cost (Keras semantics) for MI455X / gfx1250.
//
// Shapes (fixed by the reference): B=64, T=1024, C=512, L=128, S=257.
// Strategy: one workgroup (9 wave32 = 288 threads) per batch row; thread s
// owns alpha state s. Per timestep: async-DMA the next softmax row (2 KB)
// into an 8-deep LDS ring (ASYNCcnt pipelining hides memory latency), gather
// lp from LDS, banded lse3 update with alpha double-buffered in LDS, one
// split barrier per step. Memory traffic (134 MB, ~6 us at 23.3 TB/s) and
// FLOPs are trivial; the 1024-step serial dependency chain is the cost, so
// everything is organized to minimize per-step latency.
// ---------------------------------------------------------------------------

#define NEG_INF (-1e30f)
#define EPSF    (1e-7f)

constexpr int Bb    = 64;
constexpr int Tt    = 1024;
constexpr int Cc    = 512;
constexpr int Ll    = 128;
constexpr int Ss    = 2 * Ll + 1;   // 257 extended states
constexpr int BLANK = Cc - 1;       // 511
constexpr int PF    = 8;            // LDS row ring depth (async pipeline)
constexpr int NT    = 288;          // 9 wave32 per workgroup
constexpr int AW    = 292;          // alpha row width: 2 front pads + 288 + slack

#if __has_builtin(__builtin_amdgcn_global_load_async_to_lds_b128)
#define HAVE_ASYNC 1
#else
#define HAVE_ASYNC 0
#endif

#if HAVE_ASYNC
// Builtin signature (from hipcc diagnostics):
//   void __builtin_amdgcn_global_load_async_to_lds_b128(
//       v4i addrspace(1)* gsrc, v4i addrspace(3)* ldst, Imm int off, Imm int cpol)
typedef int v4i __attribute__((vector_size(16)));
#define GPTR(p) ((__attribute__((address_space(1))) v4i*)(p))
#define LPTR(p) ((__attribute__((address_space(3))) v4i*)(p))
#if __has_builtin(__builtin_amdgcn_s_wait_asynccnt)
#define WAIT_ASYNC(n) __builtin_amdgcn_s_wait_asynccnt(n)
#else
#define WAIT_ASYNC(n) asm volatile("s_wait_asynccnt " #n ::: "memory")
#endif
#endif

__global__ __launch_bounds__(NT) void ctc_scan_kernel(const int*   __restrict__ y_true,
                                                      const float* __restrict__ y_hat,
                                                      float*       __restrict__ out) {
  __shared__ float abuf[2][AW];     // double-buffered alpha, front-padded by 2
  __shared__ int   lab[Ll];         // this batch row's labels
#if HAVE_ASYNC
  __shared__ float rows[PF][Cc];    // 8-deep ring of staged softmax rows (16 KB)
#endif

  const int tid = (int)threadIdx.x;
  const int b   = (int)blockIdx.x;
  const float* yg = y_hat + (size_t)b * Tt * Cc;

#if HAVE_ASYNC
  // Prime the pipeline: DMA rows 0..PF-1 (2 KB each) into LDS.
  // 128 lanes x b128 = 2048 B per row; waves 0..3 issue, others just barrier.
  if (tid < 128) {
#pragma unroll
    for (int r = 0; r < PF; ++r) {
      __builtin_amdgcn_global_load_async_to_lds_b128(
          GPTR(yg + (size_t)r * Cc + tid * 4),
          LPTR(&rows[r][tid * 4]),
          0, 0);
    }
  }
#endif

  if (tid < Ll) lab[tid] = y_true[b * Ll + tid];
  if (tid < 2) { abuf[0][tid] = NEG_INF; abuf[1][tid] = NEG_INF; }  // front pads

#if HAVE_ASYNC
  WAIT_ASYNC(7);                    // row 0 landed (<= PF-1 outstanding)
#endif
  __syncthreads();

  // Extended-label column and skip flag for this state.
  const int s = tid;
  int  col   = BLANK;
  bool skipf = false;
  if (s < Ss) {
    if (s & 1) col = lab[s >> 1];
    skipf = (s & 1) && (s >= 3) && (lab[s >> 1] != lab[(s >> 1) - 1]);
  }

  // alpha_0: only states 0 (blank) and 1 (first label) are live.
#if HAVE_ASYNC
  const float p0 = rows[0][col];
#else
  const float p0 = yg[col];
#endif
  float alpha = (s < 2) ? __logf(p0 + EPSF) : NEG_INF;
  abuf[0][s + 2] = alpha;
  __syncthreads();

  for (int t = 1; t < Tt; ++t) {
#if HAVE_ASYNC
    const int r = t + PF - 1;       // stage row t+PF-1 into the slot row t-1 vacated
    if (r < Tt && tid < 128) {
      __builtin_amdgcn_global_load_async_to_lds_b128(
          GPTR(yg + (size_t)r * Cc + tid * 4),
          LPTR(&rows[r & (PF - 1)][tid * 4]),
          0, 0);
    }
    WAIT_ASYNC(7);                  // rows <= t are complete
    const float p = rows[t & (PF - 1)][col];
#else
    const float p = yg[(size_t)t * Cc + col];
    if (t + PF < Tt)
      __builtin_prefetch(yg + (size_t)(t + PF) * Cc + col, 0, 0);
#endif
    const float lp = __logf(p + EPSF);

    const float* ap = abuf[(t - 1) & 1];
    const float a0 = alpha;                    // state s   (kept in register)
    const float a1 = ap[s + 1];                // state s-1 (pad -> NEG_INF at s=0)
    const float a2 = skipf ? ap[s] : NEG_INF;  // state s-2 when skip allowed
    const float m  = fmaxf(a0, fmaxf(a1, a2));
    const float l  = m + __logf(__expf(a0 - m) + __expf(a1 - m) + __expf(a2 - m));
    alpha = (s < Ss) ? (lp + l) : NEG_INF;
    abuf[t & 1][s + 2] = alpha;
    __syncthreads();
  }

  if (tid == 0) {
    const float* af = abuf[(Tt - 1) & 1];
    const float x = af[Ss + 1];          // alpha[S-1]
    const float y = af[Ss];              // alpha[S-2]
    const float m = fmaxf(x, y);
    out[b] = -(m + __logf(__expf(x - m) + __expf(y - m)));
  }
}

extern "C" void kernel_launch(void* const* d_in, const int* in_sizes, int n_in,
                              void* d_out, int out_size, void* d_ws, size_t ws_size,
                              hipStream_t stream) {
  (void)in_sizes; (void)n_in; (void)d_ws; (void)ws_size; (void)out_size;
  const int*   y_true = (const int*)d_in[0];
  const float* y_hat  = (const float*)d_in[1];
  float*       out    = (float*)d_out;
  ctc_scan_kernel<<<dim3(Bb), dim3(NT), 0, stream>>>(y_true, y_hat, out);
}